// BBPMAssociativeModel_71708773974374
// MI455X (gfx1250) — compile-verified
//
#include <hip/hip_runtime.h>
#include <hip/hip_bf16.h>

// ---------------- problem constants (from reference) ----------------
#define BB        32
#define TT        2048
#define DD        512
#define VV        32000
#define NPAIRS    1023        // ts = 0,2,...,2044
#define NSLOTS_M  8191u       // NUM_SLOTS = 8192 (power of two)
#define SEED      1234u
#define GOLD      2654435769u

typedef float v2f __attribute__((ext_vector_type(2)));
typedef float v8f __attribute__((ext_vector_type(8)));

__device__ __forceinline__ unsigned mix32(unsigned h) {
    h ^= h >> 16; h *= 2246822507u;
    h ^= h >> 13; h *= 3266489909u;
    h ^= h >> 16;
    return h;
}

// ---------------------------------------------------------------------------
// Phase 1: r[b, :] = 0.25 * sum_n c[b,n] * emb[x[b,2n+1], :]
// c[b,n] = number of (j,k) pairs with probe_j(key) == probe_k(query)
// One block per batch; deterministic (fixed n order, no float atomics).
// ---------------------------------------------------------------------------
__global__ __launch_bounds__(256) void bbpm_phase1(const int* __restrict__ x,
                                                   const float* __restrict__ emb,
                                                   float* __restrict__ r) {
    const int b   = blockIdx.x;
    const int tid = threadIdx.x;
    const int* xb = x + b * TT;

    __shared__ int cnt[NPAIRS];

    // query probe slots (computed redundantly in every thread; cheap)
    const unsigned qh = mix32(((unsigned)xb[TT - 1]) ^ SEED);
    const unsigned q0 = mix32(qh + 0u * GOLD) & NSLOTS_M;
    const unsigned q1 = mix32(qh + 1u * GOLD) & NSLOTS_M;
    const unsigned q2 = mix32(qh + 2u * GOLD) & NSLOTS_M;
    const unsigned q3 = mix32(qh + 3u * GOLD) & NSLOTS_M;

    // match counts for each written pair
    for (int n = tid; n < NPAIRS; n += 256) {
        const unsigned hx = mix32(((unsigned)xb[2 * n]) ^ SEED);
        int c = 0;
#pragma unroll
        for (unsigned j = 0; j < 4; ++j) {
            const unsigned w = mix32(hx + j * GOLD) & NSLOTS_M;
            c += (w == q0) + (w == q1) + (w == q2) + (w == q3);
        }
        cnt[n] = c;
    }
    __syncthreads();

    // each thread owns 2 of the D=512 columns; accumulate in fixed n order
    const int d0 = tid * 2;
    float a0 = 0.0f, a1 = 0.0f;
    for (int n = 0; n < NPAIRS; ++n) {
        const int c = cnt[n];                 // LDS broadcast read
        if (c != 0) {                         // uniform across block
            const int tok = xb[2 * n + 1];
            const float* e = emb + (size_t)tok * DD + d0;
            const float fc = (float)c;
            a0 += fc * e[0];
            a1 += fc * e[1];
        }
    }
    r[b * DD + d0]     = 0.25f * a0;
    r[b * DD + d0 + 1] = 0.25f * a1;
}

// ---------------------------------------------------------------------------
// Phase 2: out[32, 32000] = r[32,512] @ W[32000,512]^T + bias
// fp32 WMMA 16x16x4. Block = 8 waves; wave w handles N tile
// [ (blockIdx.x*8+w)*16 , +16 ) and both M halves (M=0..15, 16..31),
// reusing each B fragment for two wmma ops.
//
// Operand layouts per CDNA5 ISA 7.12.2 (32-bit):
//   A 16x4 : lanes 0-15 -> rows M=lane, VGPR{0,1} = K{0,1};
//            lanes 16-31 -> rows M=lane-16, VGPR{0,1} = K{2,3}
//   B 4x16 : lanes 0-15 -> col N=lane, VGPR{0,1} = K{0,1};
//            lanes 16-31 -> col N=lane-16, VGPR{0,1} = K{2,3}
//   C 16x16: VGPR i, lanes 0-15 -> M=i, N=lane; lanes 16-31 -> M=i+8
// ---------------------------------------------------------------------------
__global__ __launch_bounds__(256) void bbpm_phase2(const float* __restrict__ r,
                                                   const float* __restrict__ W,
                                                   const float* __restrict__ bias,
                                                   float* __restrict__ out) {
    const int lane = threadIdx.x & 31;
    const int wave = threadIdx.x >> 5;
    const int n0   = (blockIdx.x * 8 + wave) * 16;

    const int lm = lane & 15;          // M row (A) or N col (B/C) within tile
    const int hk = (lane >> 4) * 2;    // K sub-offset: 0 for lanes 0-15, 2 for 16-31

    const float* Wp = W + (size_t)(n0 + lm) * DD + hk;   // B fragment source
    const float* r0 = r + (size_t)lm        * DD + hk;   // A, M rows 0..15
    const float* r1 = r + (size_t)(lm + 16) * DD + hk;   // A, M rows 16..31

    v8f c0 = {};
    v8f c1 = {};

    for (int k = 0; k < DD; k += 4) {
        __builtin_prefetch((const void*)(Wp + k + 256), 0, 1);  // ~1KB ahead
        const v2f bf = *(const v2f*)(Wp + k);
        const v2f a0 = *(const v2f*)(r0 + k);
        const v2f a1 = *(const v2f*)(r1 + k);
        c0 = __builtin_amdgcn_wmma_f32_16x16x4_f32(false, a0, false, bf,
                                                   (short)0, c0, false, false);
        c1 = __builtin_amdgcn_wmma_f32_16x16x4_f32(false, a1, false, bf,
                                                   (short)0, c1, false, false);
    }

    const float bv  = bias[n0 + lm];
    const int   col = n0 + lm;
    const int   mb  = (lane >> 4) * 8;   // C layout: lanes 16-31 hold M=i+8
#pragma unroll
    for (int i = 0; i < 8; ++i) {
        const int m = mb + i;
        out[(size_t)m        * VV + col] = c0[i] + bv;
        out[(size_t)(m + 16) * VV + col] = c1[i] + bv;
    }
}

// ---------------------------------------------------------------------------
extern "C" void kernel_launch(void* const* d_in, const int* in_sizes, int n_in,
                              void* d_out, int out_size, void* d_ws, size_t ws_size,
                              hipStream_t stream) {
    const int*   x    = (const int*)d_in[0];     // [32, 2048]
    const float* emb  = (const float*)d_in[1];   // [32000, 512]
    const float* W    = (const float*)d_in[2];   // [32000, 512]
    const float* bias = (const float*)d_in[3];   // [32000]
    float*       out  = (float*)d_out;           // [32, 32000]
    float*       r    = (float*)d_ws;            // [32, 512] scratch

    bbpm_phase1<<<dim3(BB), dim3(256), 0, stream>>>(x, emb, r);
    bbpm_phase2<<<dim3(VV / 128), dim3(256), 0, stream>>>(r, W, bias, out);
}